// VariationalGCNEncoder_48009144435530
// MI455X (gfx1250) — compile-verified
//
#include <hip/hip_runtime.h>
#include <math.h>

#define N_NODES 100000
#define IN_C    512
#define HID     64
#define OUT_C   32
#define N_EDGES 2000000

// padded K-major strides (mod 64 == 4 -> conflict-free b64 LDS reads)
#define W1T_STRIDE 132   // K-chunk of 128, +4 pad
#define W2T_STRIDE 68    // K of 64, +4 pad

typedef __attribute__((ext_vector_type(2))) float v2f;
typedef __attribute__((ext_vector_type(4))) float v4f;
typedef __attribute__((ext_vector_type(8))) float v8f;

static __device__ __forceinline__ v8f wmma_f32(v2f a, v2f b, v8f c) {
    // D = A(16x4) * B(4x16) + C, fp32 WMMA (VOP3P), wave32
    return __builtin_amdgcn_wmma_f32_16x16x4_f32(
        /*neg_a=*/false, a, /*neg_b=*/false, b,
        /*c_mod=*/(short)0, c, /*reuse_a=*/false, /*reuse_b=*/false);
}

// ---------------------------------------------------------------------------
// Utility kernels
// ---------------------------------------------------------------------------
__global__ __launch_bounds__(256) void zero_kernel(float* __restrict__ p, int n) {
    int i = blockIdx.x * blockDim.x + threadIdx.x;
    int stride = gridDim.x * blockDim.x;
    for (; i < n; i += stride) p[i] = 0.0f;
}

__global__ __launch_bounds__(256) void deg_kernel(const int* __restrict__ dst,
                                                  float* __restrict__ deg, int nE) {
    int e = blockIdx.x * blockDim.x + threadIdx.x;
    if (e < nE) atomicAdd(&deg[dst[e]], 1.0f);
}

__global__ __launch_bounds__(256) void dinv_kernel(float* __restrict__ deg, int n) {
    int i = blockIdx.x * blockDim.x + threadIdx.x;
    if (i < n) deg[i] = rsqrtf(deg[i] + 1.0f);  // self-loop: deg+1
}

// ---------------------------------------------------------------------------
// GEMM1: hx[N,64] = x[N,512] @ W1[512,64]   (fp32 WMMA 16x16x4)
// Block = 256 thr = 8 waves; each wave computes a 16x64 strip.
// W1 staged through LDS in K-major (transposed, padded) 128x64 chunks so a
// B fragment is one conflict-free ds_load_b64 into an even VGPR pair.
// ---------------------------------------------------------------------------
__global__ __launch_bounds__(256) void gemm1_kernel(const float* __restrict__ x,
                                                    const float* __restrict__ W1,
                                                    float* __restrict__ hx) {
    __shared__ float wldsT[HID * W1T_STRIDE];  // [col][k], 33.8 KB

    const int tid   = threadIdx.x;
    const int lane  = tid & 31;
    const int wave  = tid >> 5;
    const int col   = lane & 15;            // N within tile / M within strip
    const int khalf = (lane >> 4) << 1;     // 0 for lanes 0-15, 2 for lanes 16-31
    const int rowBase = blockIdx.x * 128 + wave * 16;

    int arow_i = rowBase + col;
    if (arow_i >= N_NODES) arow_i = N_NODES - 1;     // clamp: EXEC must stay full for WMMA
    const float* arow = x + (size_t)arow_i * IN_C;

    v8f acc[4] = {v8f{}, v8f{}, v8f{}, v8f{}};

    for (int kc = 0; kc < IN_C; kc += 128) {
        if (kc + 128 < IN_C)
            __builtin_prefetch(arow + kc + 128, 0, 1);  // global_prefetch_b8: next A chunk

        // cooperative W1 chunk load, transposed into K-major LDS:
        // thread reads 4 consecutive cols of one row, scatters with stride W1T_STRIDE
        {
            const v4f* wsrc = (const v4f*)(W1 + (size_t)kc * HID);
#pragma unroll
            for (int i = 0; i < 8; ++i) {
                int idx = tid + i * 256;       // idx over 2048 float4's
                v4f w = wsrc[idx];
                int r  = idx >> 4;             // row k within chunk
                int c0 = (idx & 15) << 2;      // starting col
                wldsT[(c0 + 0) * W1T_STRIDE + r] = w.x;
                wldsT[(c0 + 1) * W1T_STRIDE + r] = w.y;
                wldsT[(c0 + 2) * W1T_STRIDE + r] = w.z;
                wldsT[(c0 + 3) * W1T_STRIDE + r] = w.w;
            }
        }
        __syncthreads();

#pragma unroll 8
        for (int k = 0; k < 128; k += 4) {
            // A fragment: lanes0-15 hold K=k,k+1 ; lanes16-31 hold K=k+2,k+3
            v2f a = *(const v2f*)(arow + kc + k + khalf);
#pragma unroll
            for (int n = 0; n < 4; ++n) {
                // B fragment: one aligned ds_load_b64, conflict-free banks
                v2f b = *(const v2f*)(wldsT + (n * 16 + col) * W1T_STRIDE + k + khalf);
                acc[n] = wmma_f32(a, b, acc[n]);
            }
        }
        __syncthreads();
    }

    // C/D layout: VGPR r -> (M=r, N=lane) for lanes 0-15, (M=8+r, N=lane-16) for 16-31
    const int mOff = (lane >> 4) << 3;
#pragma unroll
    for (int r = 0; r < 8; ++r) {
        int rr = rowBase + mOff + r;
        if (rr < N_NODES) {
            float* orow = hx + (size_t)rr * HID + col;
            orow[0]  = acc[0][r];
            orow[16] = acc[1][r];
            orow[32] = acc[2][r];
            orow[48] = acc[3][r];
        }
    }
}

// ---------------------------------------------------------------------------
// Edge aggregation: agg[dst] += feat[src] * dinv[src]*dinv[dst]
// One edge per wave; lane l handles features 2l,2l+1 (coalesced 256B gather).
// Atomics land in L2 (agg buffer is 25.6 MB < 192 MB L2).
// ---------------------------------------------------------------------------
__global__ __launch_bounds__(256) void agg_kernel(const int* __restrict__ src,
                                                  const int* __restrict__ dst,
                                                  const float* __restrict__ dinv,
                                                  const float* __restrict__ feat,
                                                  float* __restrict__ agg) {
    const int lane = threadIdx.x & 31;
    const int e = (blockIdx.x * blockDim.x + threadIdx.x) >> 5;
    if (e >= N_EDGES) return;
    const int s = src[e];
    const int d = dst[e];
    const float norm = dinv[s] * dinv[d];
    const float2 v = ((const float2*)(feat + (size_t)s * HID))[lane];
    float* ap = agg + (size_t)d * HID + lane * 2;
    atomicAdd(ap,     v.x * norm);
    atomicAdd(ap + 1, v.y * norm);
}

// h = relu(agg + dinv^2 * hx + b1)   (written in place over agg)
__global__ __launch_bounds__(256) void finalize1_kernel(float* __restrict__ h_agg,
                                                        const float* __restrict__ hx,
                                                        const float* __restrict__ dinv,
                                                        const float* __restrict__ b1) {
    int i = blockIdx.x * blockDim.x + threadIdx.x;
    if (i >= N_NODES * HID) return;
    int node = i >> 6;
    int f = i & 63;
    float di = dinv[node];
    float v = h_agg[i] + di * di * hx[i] + b1[f];
    h_agg[i] = fmaxf(v, 0.0f);
}

// P = agg + dinv^2 * h   (in place over agg)
__global__ __launch_bounds__(256) void finalizeP_kernel(float* __restrict__ p_agg,
                                                        const float* __restrict__ h,
                                                        const float* __restrict__ dinv) {
    int i = blockIdx.x * blockDim.x + threadIdx.x;
    if (i >= N_NODES * HID) return;
    int node = i >> 6;
    float di = dinv[node];
    p_agg[i] = p_agg[i] + di * di * h[i];
}

// ---------------------------------------------------------------------------
// GEMM2: mu = P @ W_mu + b_mu ; logstd = P @ W_ls + b_ls   (K=64, N=32 each)
// Weights transposed K-major in LDS (stride 68, conflict-free b64 reads).
// ---------------------------------------------------------------------------
__global__ __launch_bounds__(256) void gemm2_kernel(const float* __restrict__ P,
                                                    const float* __restrict__ Wmu,
                                                    const float* __restrict__ bmu,
                                                    const float* __restrict__ Wls,
                                                    const float* __restrict__ bls,
                                                    float* __restrict__ out) {
    __shared__ float wmT[OUT_C * W2T_STRIDE];  // [col][k]
    __shared__ float wlT[OUT_C * W2T_STRIDE];
    __shared__ float bm[OUT_C];
    __shared__ float bl[OUT_C];

    const int tid = threadIdx.x;
    for (int i = tid; i < HID * OUT_C; i += 256) {
        int k = i >> 5;          // row (K)
        int c = i & 31;          // col (N)
        wmT[c * W2T_STRIDE + k] = Wmu[i];
        wlT[c * W2T_STRIDE + k] = Wls[i];
    }
    if (tid < OUT_C) { bm[tid] = bmu[tid]; bl[tid] = bls[tid]; }
    __syncthreads();

    const int lane  = tid & 31;
    const int wave  = tid >> 5;
    const int col   = lane & 15;
    const int khalf = (lane >> 4) << 1;
    const int rowBase = blockIdx.x * 128 + wave * 16;

    int arow_i = rowBase + col;
    if (arow_i >= N_NODES) arow_i = N_NODES - 1;
    const float* arow = P + (size_t)arow_i * HID;

    v8f am0{}, am1{}, al0{}, al1{};
#pragma unroll
    for (int k = 0; k < HID; k += 4) {
        v2f a = *(const v2f*)(arow + k + khalf);
        const int kk = k + khalf;
        v2f b;
        b = *(const v2f*)(wmT + col * W2T_STRIDE + kk);
        am0 = wmma_f32(a, b, am0);
        b = *(const v2f*)(wmT + (16 + col) * W2T_STRIDE + kk);
        am1 = wmma_f32(a, b, am1);
        b = *(const v2f*)(wlT + col * W2T_STRIDE + kk);
        al0 = wmma_f32(a, b, al0);
        b = *(const v2f*)(wlT + (16 + col) * W2T_STRIDE + kk);
        al1 = wmma_f32(a, b, al1);
    }

    const size_t LS_OFF = (size_t)N_NODES * OUT_C;
    const int mOff = (lane >> 4) << 3;
#pragma unroll
    for (int r = 0; r < 8; ++r) {
        int rr = rowBase + mOff + r;
        if (rr < N_NODES) {
            size_t o = (size_t)rr * OUT_C + col;
            out[o]               = am0[r] + bm[col];
            out[o + 16]          = am1[r] + bm[16 + col];
            out[LS_OFF + o]      = al0[r] + bl[col];
            out[LS_OFF + o + 16] = al1[r] + bl[16 + col];
        }
    }
}

// ---------------------------------------------------------------------------
extern "C" void kernel_launch(void* const* d_in, const int* in_sizes, int n_in,
                              void* d_out, int out_size, void* d_ws, size_t ws_size,
                              hipStream_t stream) {
    const float* x   = (const float*)d_in[0];
    const int*   ei  = (const int*)d_in[1];
    const float* W1  = (const float*)d_in[2];
    const float* b1  = (const float*)d_in[3];
    const float* Wmu = (const float*)d_in[4];
    const float* bmu = (const float*)d_in[5];
    const float* Wls = (const float*)d_in[6];
    const float* bls = (const float*)d_in[7];
    float* out = (float*)d_out;

    const int* src = ei;            // edge_index[0]
    const int* dst = ei + N_EDGES;  // edge_index[1]

    // workspace layout (floats): hx | h | P | dinv   (~77 MB total)
    float* ws   = (float*)d_ws;
    float* hx   = ws;
    float* h    = hx + (size_t)N_NODES * HID;
    float* Pb   = h  + (size_t)N_NODES * HID;
    float* dinv = Pb + (size_t)N_NODES * HID;

    const int feTotal = N_NODES * HID;  // 6.4M

    // zero h, P, deg (contiguous region)
    zero_kernel<<<4096, 256, 0, stream>>>(h, 2 * feTotal + N_NODES);
    // degree (accumulated into dinv buffer), then dinv = rsqrt(deg+1)
    deg_kernel<<<(N_EDGES + 255) / 256, 256, 0, stream>>>(dst, dinv, N_EDGES);
    dinv_kernel<<<(N_NODES + 255) / 256, 256, 0, stream>>>(dinv, N_NODES);
    // layer 1 transform
    gemm1_kernel<<<(N_NODES + 127) / 128, 256, 0, stream>>>(x, W1, hx);
    // layer 1 aggregation + self-loop + bias + relu
    agg_kernel<<<N_EDGES / 8, 256, 0, stream>>>(src, dst, dinv, hx, h);
    finalize1_kernel<<<(feTotal + 255) / 256, 256, 0, stream>>>(h, hx, dinv, b1);
    // shared aggregation P = A_hat * h  (mu and logstd both consume it)
    agg_kernel<<<N_EDGES / 8, 256, 0, stream>>>(src, dst, dinv, h, Pb);
    finalizeP_kernel<<<(feTotal + 255) / 256, 256, 0, stream>>>(Pb, h, dinv);
    // fused small GEMMs: mu and logstd
    gemm2_kernel<<<(N_NODES + 127) / 128, 256, 0, stream>>>(Pb, Wmu, bmu, Wls, bls, out);
}